// GraphStructuralLayer_3633542332533
// MI455X (gfx1250) — compile-verified
//
#include <hip/hip_runtime.h>

typedef __attribute__((ext_vector_type(2))) float v2f;
typedef __attribute__((ext_vector_type(8))) float v8f;

#define N_NODES 50000
#define N_EDGES 500000
#define HID 128
#define HEADS 8
#define HDIM 16

// ---------------------------------------------------------------------------
// Zero-fill helper (agg / max / sum regions must be reset every call).
// ---------------------------------------------------------------------------
__global__ void k_zero(float* __restrict__ p, int n) {
    int i = blockIdx.x * blockDim.x + threadIdx.x;
    if (i < n) p[i] = 0.0f;
}

// ---------------------------------------------------------------------------
// Fused 4-way GEMM: O_t[N x 128] = X[N x 128] * W_t[128 x 128] + b_t,
// t = Q,K,V,skip.  fp32 WMMA (v_wmma_f32_16x16x4_f32).
//
// 256 threads = 8 waves. Each wave owns a 16-row stripe of A, loaded ONCE
// into 64 VGPRs/lane (32 k-pairs), then reused across all 4 weight matrices.
// Each W_t is staged into LDS in a pair-interleaved layout so every B
// fragment is a single 8B-aligned ds_load_b64 with an immediate offset:
//     sW[(k>>1)*256 + col*2 + (k&1)]  ==  W[k][col]
// ---------------------------------------------------------------------------
__global__ void __launch_bounds__(256) k_gemm_qkvs(
        const float* __restrict__ X,
        const float* __restrict__ W0, const float* __restrict__ B0, float* __restrict__ O0,
        const float* __restrict__ W1, const float* __restrict__ B1, float* __restrict__ O1,
        const float* __restrict__ W2, const float* __restrict__ B2, float* __restrict__ O2,
        const float* __restrict__ W3, const float* __restrict__ B3, float* __restrict__ O3,
        int nrows) {
    __shared__ float sW[HID * HID];  // 64 KB (WGP has 320 KB)

    const float* Wt[4] = {W0, W1, W2, W3};
    const float* Bt[4] = {B0, B1, B2, B3};
    float*       Ot[4] = {O0, O1, O2, O3};

    const int wave = threadIdx.x >> 5;
    const int lane = threadIdx.x & 31;
    const int rowTile = blockIdx.x * 8 + wave;
    const bool active = (rowTile * 16) < nrows;  // wave-uniform

    const int row0 = rowTile << 4;
    const int m = lane & 15;     // A row / B,D column within tile
    const int half = lane >> 4;  // K-pair selector

    // ---- Load the whole A stripe for this wave into registers (once). ----
    v2f a[32];
    if (active) {
        const float* arow = X + (size_t)(row0 + m) * HID + 2 * half;
#pragma unroll
        for (int kk = 0; kk < 32; kk++) a[kk] = *(const v2f*)(arow + kk * 4);
    }

#pragma unroll 1
    for (int t = 0; t < 4; t++) {
        __syncthreads();  // previous stage's LDS readers are done
        // Cooperative pair-interleaved stage of W_t into LDS.
        const float* w = Wt[t];
        for (int i = threadIdx.x; i < HID * HID; i += 256) {
            const int k = i >> 7, col = i & 127;
            sW[(k >> 1) * 256 + col * 2 + (k & 1)] = w[i];
        }
        __syncthreads();

        if (active) {
            v8f acc[8] = {};
            // base: pair-row = 2*kk + half, lane column = m
            const float* bbase = sW + half * 256 + m * 2;
#pragma unroll
            for (int kk = 0; kk < 32; kk++) {
                const float* brow = bbase + kk * 512;  // (2*kk)*256 floats
#pragma unroll
                for (int ct = 0; ct < 8; ct++) {
                    v2f b = *(const v2f*)(brow + ct * 32);  // ds_load_b64
                    acc[ct] = __builtin_amdgcn_wmma_f32_16x16x4_f32(
                        false, a[kk], false, b, (short)0, acc[ct], false, false);
                }
            }
            // Epilogue: D VGPR r holds row r (lanes 0-15) / row r+8 (16-31).
            const float* bias = Bt[t];
            float* Out = Ot[t];
#pragma unroll
            for (int ct = 0; ct < 8; ct++) {
                const int col = ct * 16 + m;
                const float bv = bias[col];
#pragma unroll
                for (int r = 0; r < 8; r++) {
                    const int row = row0 + r + 8 * half;
                    Out[(size_t)row * HID + col] = acc[ct][r] + bv;
                }
            }
        }
    }
}

// Monotonic float<->uint order mapping for atomic max on f32 scores.
__device__ __forceinline__ unsigned enc_ord(float f) {
    unsigned u = __float_as_uint(f);
    return (u & 0x80000000u) ? ~u : (u | 0x80000000u);
}
__device__ __forceinline__ float dec_ord(unsigned u) {
    return __uint_as_float((u & 0x80000000u) ? (u ^ 0x80000000u) : ~u);
}

// ---------------------------------------------------------------------------
// Per-(edge,head) attention score + running segment max (global_atomic_max_u32).
// q/k gather tables (~51 MB) are L2-resident (192 MB).
// ---------------------------------------------------------------------------
__global__ void k_score(const float* __restrict__ q, const float* __restrict__ k,
                        const int* __restrict__ src, const int* __restrict__ dst,
                        float* __restrict__ score, unsigned* __restrict__ mmax) {
    int idx = blockIdx.x * 256 + threadIdx.x;
    if (idx >= N_EDGES * HEADS) return;
    const int e = idx >> 3, h = idx & 7;
    const int sN = src[e], dN = dst[e];
    const float4* qp = (const float4*)(q + (size_t)dN * HID + h * HDIM);
    const float4* kp = (const float4*)(k + (size_t)sN * HID + h * HDIM);
    float acc = 0.0f;
#pragma unroll
    for (int j = 0; j < 4; j++) {
        float4 a = qp[j], b = kp[j];
        acc += a.x * b.x + a.y * b.y + a.z * b.z + a.w * b.w;
    }
    acc *= 0.25f;  // 1/sqrt(16)
    score[idx] = acc;
    atomicMax(&mmax[(size_t)dN * HEADS + h], enc_ord(acc));
}

// ---------------------------------------------------------------------------
// e = exp(score - max[dst]); segment sum via global_atomic_add_f32.
// ---------------------------------------------------------------------------
__global__ void k_expsum(const int* __restrict__ dst,
                         const unsigned* __restrict__ mmax,
                         float* __restrict__ score, float* __restrict__ ssum) {
    int idx = blockIdx.x * 256 + threadIdx.x;
    if (idx >= N_EDGES * HEADS) return;
    const int e = idx >> 3, h = idx & 7;
    const int dN = dst[e];
    const float mval = dec_ord(mmax[(size_t)dN * HEADS + h]);
    const float ee = __expf(score[idx] - mval);
    score[idx] = ee;
    atomicAdd(&ssum[(size_t)dN * HEADS + h], ee);
}

// ---------------------------------------------------------------------------
// agg[dst] += alpha * v[src]  (16 f32 atomics per (edge,head))
// ---------------------------------------------------------------------------
__global__ void k_agg(const int* __restrict__ src, const int* __restrict__ dst,
                      const float* __restrict__ score, const float* __restrict__ ssum,
                      const float* __restrict__ v, float* __restrict__ agg) {
    int idx = blockIdx.x * 256 + threadIdx.x;
    if (idx >= N_EDGES * HEADS) return;
    const int e = idx >> 3, h = idx & 7;
    const int sN = src[e], dN = dst[e];
    const float alpha = score[idx] / (ssum[(size_t)dN * HEADS + h] + 1e-16f);
    const float4* vp = (const float4*)(v + (size_t)sN * HID + h * HDIM);
    float* ap = agg + (size_t)dN * HID + h * HDIM;
#pragma unroll
    for (int j = 0; j < 4; j++) {
        float4 vv = vp[j];
        atomicAdd(ap + 4 * j + 0, alpha * vv.x);
        atomicAdd(ap + 4 * j + 1, alpha * vv.y);
        atomicAdd(ap + 4 * j + 2, alpha * vv.z);
        atomicAdd(ap + 4 * j + 3, alpha * vv.w);
    }
}

// ---------------------------------------------------------------------------
// out = prelu(agg + skip [+ resid])
// ---------------------------------------------------------------------------
__global__ void k_finalize(const float* __restrict__ agg, const float* __restrict__ skip,
                           const float* __restrict__ resid, const float* __restrict__ prelu_w,
                           float* __restrict__ out, int n, int do_resid) {
    int i = blockIdx.x * 256 + threadIdx.x;
    if (i >= n) return;
    float o = agg[i] + skip[i];
    if (do_resid) o += resid[i];
    const float a = prelu_w[0];
    out[i] = fmaxf(o, 0.0f) + a * fminf(o, 0.0f);
}

// ---------------------------------------------------------------------------
extern "C" void kernel_launch(void* const* d_in, const int* in_sizes, int n_in,
                              void* d_out, int out_size, void* d_ws, size_t ws_size,
                              hipStream_t stream) {
    const float* x       = (const float*)d_in[0];
    const int*   ei      = (const int*)d_in[1];
    const float* prelu_w = (const float*)d_in[2];
    const float* Wl[2][4];
    const float* bl[2][4];
    for (int L = 0; L < 2; L++)
        for (int t = 0; t < 4; t++) {
            Wl[L][t] = (const float*)d_in[3 + L * 8 + t * 2];
            bl[L][t] = (const float*)d_in[3 + L * 8 + t * 2 + 1];
        }
    const int* src = ei;
    const int* dst = ei + N_EDGES;

    const size_t NF = (size_t)N_NODES * HID;   // 6.4M floats
    const size_t E8 = (size_t)N_EDGES * HEADS; // 4.0M floats
    const size_t NH = (size_t)N_NODES * HEADS; // 0.4M floats

    float* ws    = (float*)d_ws;
    float* q     = ws;            // NF
    float* kbuf  = q + NF;        // NF
    float* vbuf  = kbuf + NF;     // NF
    float* skip  = vbuf + NF;     // NF
    float* hbuf  = skip + NF;     // NF  (layer-1 output)
    float* score = hbuf + NF;     // E8
    float* agg   = score + E8;    // NF  -- zeroed each layer
    unsigned* mmax = (unsigned*)(agg + NF); // NH -- zeroed (== encoded -inf)
    float* ssum  = (float*)(mmax + NH);     // NH -- zeroed

    const int zeroN    = (int)(NF + NH + NH);
    const int gemmGrid = (N_NODES / 16 + 7) / 8;  // 3125 row tiles / 8 waves
    const int ehGrid   = (int)((E8 + 255) / 256);
    const int nfGrid   = (int)((NF + 255) / 256);

    for (int L = 0; L < 2; L++) {
        const float* Xin  = (L == 0) ? x : hbuf;
        float*       Yout = (L == 0) ? hbuf : (float*)d_out;

        k_zero<<<(zeroN + 255) / 256, 256, 0, stream>>>(agg, zeroN);

        k_gemm_qkvs<<<gemmGrid, 256, 0, stream>>>(
            Xin,
            Wl[L][0], bl[L][0], q,
            Wl[L][1], bl[L][1], kbuf,
            Wl[L][2], bl[L][2], vbuf,
            Wl[L][3], bl[L][3], skip,
            N_NODES);

        k_score <<<ehGrid, 256, 0, stream>>>(q, kbuf, src, dst, score, mmax);
        k_expsum<<<ehGrid, 256, 0, stream>>>(dst, mmax, score, ssum);
        k_agg   <<<ehGrid, 256, 0, stream>>>(src, dst, score, ssum, vbuf, agg);

        k_finalize<<<nfGrid, 256, 0, stream>>>(agg, skip, x, prelu_w, Yout,
                                               (int)NF, /*do_resid=*/L == 1);
    }
}